// MotionSaliencySelector_49392123904659
// MI455X (gfx1250) — compile-verified
//
#include <hip/hip_runtime.h>
#include <stdint.h>
#include <math.h>

#define TT 1024
#define DD 263
#define TILE_T 128
#define TILE_ROWS 131           // 128 + 3-row halo for t-1, t-2, t-3
#define TILE_W 67               // features 0..66 are all we need
#define NTHREADS 128

typedef unsigned int u32x4 __attribute__((ext_vector_type(4)));
typedef int          i32x8 __attribute__((ext_vector_type(8)));
typedef int          i32x4 __attribute__((ext_vector_type(4)));

// per-joint weight = group_weight / group_size
// groups: RS{0,3,6,9,12,15}w=1.6  FEET{7,8,10,11}w=1.4  HANDS{13,14,16..21}w=1.2  OTHER{1,2,4,5}w=0.8
__device__ __constant__ float JW[22] = {
  1.6f/6.f, 0.2f, 0.2f, 1.6f/6.f, 0.2f, 0.2f, 1.6f/6.f,
  0.35f, 0.35f, 1.6f/6.f, 0.35f, 0.35f, 1.6f/6.f,
  0.15f, 0.15f, 1.6f/6.f, 0.15f, 0.15f, 0.15f, 0.15f, 0.15f, 0.15f };

// ---- Tensor Data Mover: 2D tile (tile_w x rows) from a row-major (.., DD) tensor into LDS.
// D# packed per CDNA5 ISA 8.3/8.4: group0 {count=1, lds_addr, global_addr, type=2},
// group1 {data_size=4B, tensor_dim0=DD, tensor_dim1, tile_dim0, tile_dim1, dim0_stride=DD}.
__device__ inline void tdm_load_2d(const void* gptr, void* lptr,
                                   unsigned tile_w, unsigned rows, unsigned tdim1) {
  unsigned long long ga = (unsigned long long)(uintptr_t)gptr;
  unsigned lds = (unsigned)(uintptr_t)lptr;  // low 32 bits of generic ptr == LDS byte offset
  u32x4 g0 = { 1u,                                           // count=1, user mode
               lds,                                          // lds_addr
               (unsigned)(ga & 0xFFFFFFFFull),               // global_addr[31:0]
               (unsigned)((ga >> 32) & 0x01FFFFFFull) | 0x80000000u }; // addr[56:32] | type=2
  i32x8 g1 = {
    (int)(2u << 16),                                         // data_size = 2 (4 bytes)
    (int)((DD & 0xFFFFu) << 16),                             // tensor_dim0[15:0]
    (int)((tdim1 & 0xFFFFu) << 16),                          // tensor_dim0[31:16]=0 | tensor_dim1[15:0]
    (int)(((tdim1 >> 16) & 0xFFFFu) | (tile_w << 16)),       // tensor_dim1[31:16] | tile_dim0
    (int)(rows & 0xFFFFu),                                   // tile_dim1 (tile_dim2=0)
    (int)DD,                                                 // tensor_dim0_stride[31:0]
    0, 0 };
  i32x4 z4 = {0, 0, 0, 0};                                   // groups 2/3 unused (2D tensor)
  i32x8 z8 = {0, 0, 0, 0, 0, 0, 0, 0};                       // extra group (unused)
  __builtin_amdgcn_tensor_load_to_lds(g0, g1, z4, z4, z8, 0);
}

__device__ inline float sn3(float x, float y, float z) {
  return sqrtf(x*x + y*y + z*z + 1e-12f);
}

// vel/acc/jrk energies from positions at t, t-1, t-2, t-3 (clamped)
__device__ inline float energy3(const float* px, const float* py, const float* pz) {
  float v0x = px[0]-px[1], v0y = py[0]-py[1], v0z = pz[0]-pz[1];
  float v1x = px[1]-px[2], v1y = py[1]-py[2], v1z = pz[1]-pz[2];
  float v2x = px[2]-px[3], v2y = py[2]-py[3], v2z = pz[2]-pz[3];
  float ax = v0x - v1x, ay = v0y - v1y, az = v0z - v1z;
  float jx = ax - (v1x - v2x), jy = ay - (v1y - v2y), jz = az - (v1z - v2z);
  return sn3(v0x,v0y,v0z) + 0.6f*sn3(ax,ay,az) + 0.35f*sn3(jx,jy,jz);
}

__global__ __launch_bounds__(NTHREADS)
void motion_saliency_kernel(const float* __restrict__ motion,
                            const unsigned char* __restrict__ vmask,
                            float* __restrict__ out) {
  const int b   = blockIdx.x;
  const int tid = threadIdx.x;

  __shared__ float k1s[TT], k2s[TT], rxs[TT], rzs[TT], rys[TT], Ps[TT];
  __shared__ float tileBuf[TILE_ROWS * TILE_W];   // also reused as sort buffer
  __shared__ float red[NTHREADS], red2[NTHREADS];
  __shared__ unsigned char mbuf[TT];

  // ---------------- mask load + count ----------------
  int cnt = 0;
  for (int kk = 0; kk < 8; ++kk) {
    int t = kk * NTHREADS + tid;                  // coalesced
    unsigned char m = vmask[(size_t)b * TT + t];
    mbuf[t] = m;
    cnt += (m != 0);
  }
  red[tid] = (float)cnt;
  __syncthreads();
  for (int off = NTHREADS/2; off > 0; off >>= 1) {
    if (tid < off) red[tid] += red[tid + off];
    __syncthreads();
  }
  const int   n_valid = (int)red[0];
  const float hvf  = (n_valid > 0) ? 1.0f : 0.0f;
  const int   nmax = (n_valid > 1) ? n_valid : 1;
  const int   last = nmax - 1;
  const float nf   = (float)nmax;
  __syncthreads();                                 // red reused below

  // ---------------- TDM: root features (cols 0..3) for all t ----------------
  if (tid < 32) {
    tdm_load_2d(motion + (size_t)b * TT * DD, tileBuf, 4u, (unsigned)TT, (unsigned)TT);
    __builtin_amdgcn_s_wait_tensorcnt(0);
  }
  __syncthreads();

  // ---------------- pass 1: exclusive scan of rot_vel -> ang -> (k1,k2) ----------------
  float seg[8];
  {
    float run = 0.f;
    for (int kk = 0; kk < 8; ++kk) {
      int t = tid * 8 + kk;
      seg[kk] = run;                               // exclusive within segment
      run += tileBuf[t * 4 + 0];
    }
    red[tid] = run;
  }
  __syncthreads();
  for (int off = 1; off < NTHREADS; off <<= 1) {
    float v = red[tid] + ((tid >= off) ? red[tid - off] : 0.f);
    __syncthreads();
    red[tid] = v;
    __syncthreads();
  }
  float base = (tid > 0) ? red[tid - 1] : 0.f;
  float mk1[8], mk2[8];
  for (int kk = 0; kk < 8; ++kk) {
    int t = tid * 8 + kk;
    float a = base + seg[kk];
    float s = sinf(a), c = cosf(a);
    float K1 = 1.f - 2.f * s * s;                  // qrot(qinv(q), v) about Y
    float K2 = 2.f * s * c;
    k1s[t] = K1;  k2s[t] = K2;
    mk1[kk] = K1; mk2[kk] = K2;
    rys[t] = tileBuf[t * 4 + 3];                   // root y = data[...,3]
  }
  __syncthreads();

  // ---------------- pass 1b: inclusive scan of rotated steps -> r_pos x/z ----------------
  float segx[8], segz[8];
  {
    float rx = 0.f, rz = 0.f;
    for (int kk = 0; kk < 8; ++kk) {
      int t = tid * 8 + kk;
      float vx = 0.f, vz = 0.f;
      if (t >= 1) { vx = tileBuf[(t - 1) * 4 + 1]; vz = tileBuf[(t - 1) * 4 + 2]; }
      rx += vx * mk1[kk] - mk2[kk] * vz;
      rz += vz * mk1[kk] + mk2[kk] * vx;
      segx[kk] = rx; segz[kk] = rz;                // inclusive within segment
    }
    red[tid] = rx; red2[tid] = rz;
  }
  __syncthreads();
  for (int off = 1; off < NTHREADS; off <<= 1) {
    float v  = red[tid]  + ((tid >= off) ? red[tid - off]  : 0.f);
    float v2 = red2[tid] + ((tid >= off) ? red2[tid - off] : 0.f);
    __syncthreads();
    red[tid] = v; red2[tid] = v2;
    __syncthreads();
  }
  float bx = (tid > 0) ? red[tid - 1]  : 0.f;
  float bz = (tid > 0) ? red2[tid - 1] : 0.f;
  for (int kk = 0; kk < 8; ++kk) {
    int t = tid * 8 + kk;
    rxs[t] = bx + segx[kk];
    rzs[t] = bz + segz[kk];
  }
  __syncthreads();

  // ---------------- pass 2: tiled saliency (TDM-staged joints, cols 0..66) ----------------
  for (int t0 = 0; t0 < TT; t0 += TILE_T) {
    int rstart = t0 - 3; if (rstart < 0) rstart = 0;
    __syncthreads();                               // previous tile fully consumed
    if (tid < 32) {
      tdm_load_2d(motion + ((size_t)b * TT + rstart) * DD, tileBuf,
                  (unsigned)TILE_W, (unsigned)TILE_ROWS, (unsigned)TILE_ROWS);
      __builtin_amdgcn_s_wait_tensorcnt(0);
    }
    __syncthreads();

    const int t = t0 + tid;
    float K1[4], K2[4], RX[4], RZ[4], RY[4];
    int row[4];
    for (int k = 0; k < 4; ++k) {
      int tp = t - k; if (tp < 0) tp = 0;          // prepend-clamp semantics
      K1[k] = k1s[tp]; K2[k] = k2s[tp];
      RX[k] = rxs[tp]; RZ[k] = rzs[tp]; RY[k] = rys[tp];
      row[k] = (tp - rstart) * TILE_W;
    }
    float e = 0.f;
    {   // joint 0 = root position (rx, ry, rz)
      float px[4], py[4], pz[4];
      for (int k = 0; k < 4; ++k) { px[k] = RX[k]; py[k] = RY[k]; pz[k] = RZ[k]; }
      e += JW[0] * energy3(px, py, pz);
    }
    for (int j = 1; j < 22; ++j) {
      const int cb = 4 + 3 * (j - 1);
      float px[4], py[4], pz[4];
      for (int k = 0; k < 4; ++k) {
        float jx = tileBuf[row[k] + cb + 0];
        float jy = tileBuf[row[k] + cb + 1];
        float jz = tileBuf[row[k] + cb + 2];
        px[k] = jx * K1[k] - K2[k] * jz + RX[k];
        py[k] = jy;
        pz[k] = jz * K1[k] + K2[k] * jx + RZ[k];
      }
      e += JW[j] * energy3(px, py, pz);
    }
    // turning term from root xz
    float pvx = RX[0] - RX[1], pvz = RZ[0] - RZ[1];
    float qvx = RX[1] - RX[2], qvz = RZ[1] - RZ[2];
    bool z0 = (fabsf(pvx) < 1e-8f) && (fabsf(pvz) < 1e-8f);
    bool z1 = (fabsf(qvx) < 1e-8f) && (fabsf(qvz) < 1e-8f);
    float h0 = z0 ? 0.f : atan2f(pvz, pvx);
    float h1 = z1 ? 0.f : atan2f(qvz, qvx);
    float hd = h0 - h1;
    float hw = atan2f(sinf(hd), cosf(hd));         // wrap to [-pi, pi]
    float turn = fabsf(hw) * sqrtf(pvx * pvx + pvz * pvz + 1e-12f);
    Ps[t] = e + 1.6f * turn;
  }
  __syncthreads();

  // ---------------- normalize by row max ----------------
  float msal[8];
  float mx = 0.f;
  for (int kk = 0; kk < 8; ++kk) {
    int t = kk * NTHREADS + tid;
    float mf = mbuf[t] ? 1.f : 0.f;
    float v = Ps[t] * mf;
    msal[kk] = v;
    mx = fmaxf(mx, v);
  }
  red[tid] = mx;
  __syncthreads();
  for (int off = NTHREADS/2; off > 0; off >>= 1) {
    if (tid < off) red[tid] = fmaxf(red[tid], red[tid + off]);
    __syncthreads();
  }
  const float mnorm = fmaxf(red[0], 1e-6f);
  __syncthreads();
  for (int kk = 0; kk < 8; ++kk) {
    int t = kk * NTHREADS + tid;
    float mf = mbuf[t] ? 1.f : 0.f;
    float v = fminf(fmaxf(msal[kk] / mnorm, 0.f), 1.f) * mf;
    Ps[t] = v;
  }
  __syncthreads();

  // ---------------- local-max filter + endpoint #1 ----------------
  float pr[8];
  for (int kk = 0; kk < 8; ++kk) {
    int t = kk * NTHREADS + tid;
    float v = Ps[t];
    float lm = v;
    for (int d = -2; d <= 2; ++d) {
      int u = t + d;
      if (u >= 0 && u < TT) lm = fmaxf(lm, Ps[u]);
    }
    float mf = mbuf[t] ? 1.f : 0.f;
    float p = (v >= lm - 1e-6f) ? v : 0.f;
    p *= mf;
    p = fminf(fmaxf(p, 0.f), 1.f) * mf;            // clip in reference()
    if (t == 0 || t == last) p = fmaxf(p, hvf);    // endpoint
    p *= mf;
    pr[kk] = p;
  }
  __syncthreads();
  float ssum = 0.f;
  for (int kk = 0; kk < 8; ++kk) {
    int t = kk * NTHREADS + tid;
    Ps[t] = pr[kk];
    ssum += pr[kk];
  }
  __syncthreads();

  // ---------------- activity -> quantile position ----------------
  red[tid] = ssum;
  __syncthreads();
  for (int off = NTHREADS/2; off > 0; off >>= 1) {
    if (tid < off) red[tid] += red[tid + off];
    __syncthreads();
  }
  const float activity = red[0] / nf;
  float q = 0.9f - 0.1f * activity;                // 0.85 + 0.1*0.5 - 0.1*act
  q = fminf(fmaxf(q, 0.8f), 0.95f);
  const float pos = q * (nf - 1.f);
  const int lo = (int)floorf(pos);
  const int hi = (int)ceilf(pos);
  __syncthreads();

  // ---------------- bitonic sort (masked probs, invalid->2.0) ----------------
  float* sb = tileBuf;                              // reuse tile LDS as sort buffer
  for (int kk = 0; kk < 8; ++kk) {
    int t = kk * NTHREADS + tid;
    sb[t] = mbuf[t] ? Ps[t] : 2.0f;
  }
  __syncthreads();
  for (unsigned k = 2; k <= (unsigned)TT; k <<= 1) {
    for (unsigned j = k >> 1; j > 0; j >>= 1) {
      for (unsigned i = tid; i < (unsigned)TT; i += NTHREADS) {
        unsigned ixj = i ^ j;
        if (ixj > i) {
          float a = sb[i], c = sb[ixj];
          bool swap = ((i & k) == 0) ? (a > c) : (a < c);
          if (swap) { sb[i] = c; sb[ixj] = a; }
        }
      }
      __syncthreads();
    }
  }
  const float vlo = sb[lo], vhi = sb[hi];
  const float cut = vlo + (pos - floorf(pos)) * (vhi - vlo);

  // ---------------- hard mask + endpoint #2, write outputs ----------------
  const size_t ob   = (size_t)b * TT;
  const size_t ost  = (size_t)gridDim.x * TT;       // st follows probs in d_out
  for (int kk = 0; kk < 8; ++kk) {
    int t = kk * NTHREADS + tid;
    float mf = mbuf[t] ? 1.f : 0.f;
    float P  = Ps[t];
    float h;
    if      (n_valid > 2) h = (P >= cut) ? 1.f : 0.f;
    else if (n_valid > 0) h = mf;
    else                  h = 0.f;
    if (t == 0 || t == last) { h = fmaxf(h, hvf); P = fmaxf(P, hvf); }
    h *= mf; P *= mf;
    out[ob + t]       = P;                          // probs
    out[ost + ob + t] = h;                          // st == hard at runtime
  }
}

extern "C" void kernel_launch(void* const* d_in, const int* in_sizes, int n_in,
                              void* d_out, int out_size, void* d_ws, size_t ws_size,
                              hipStream_t stream) {
  (void)n_in; (void)out_size; (void)d_ws; (void)ws_size;
  const float* motion        = (const float*)d_in[0];
  const unsigned char* mask  = (const unsigned char*)d_in[1];  // jnp bool = 1 byte/elem
  float* out                 = (float*)d_out;
  const int B = in_sizes[0] / (TT * DD);
  hipLaunchKernelGGL(motion_saliency_kernel, dim3(B), dim3(NTHREADS), 0, stream,
                     motion, mask, out);
}